// YoloV7PoseLoss_29446295781509
// MI455X (gfx1250) — compile-verified
//
#include <hip/hip_runtime.h>
#include <math.h>

// ---------------- problem constants (from reference) ----------------
#define NA_   3
#define NT_   300
#define NKPT_ 17
#define BS_   16
#define NO_   57          // 5 + NC + 3*NKPT
#define D0_   40          // target row width: b,c,x,y,w,h + 34 kpt coords
#define ROWS_ (5 * NA_ * NT_)   // 4500 candidate rows per level

// workspace layout (floats):
//  [0..4]   level0 sums: cnt, s_box, s_kptv, s_kpt, n_pos
//  [5..9]   level1 sums
//  [10..14] level2 sums
//  [15..17] obj BCE sums per level
//  [32..]   tobj grids: 16*3*80*80 + 16*3*40*40 + 16*3*20*20 = 403200 floats
#define WS_HDR_ 32

typedef float v2f __attribute__((ext_vector_type(2)));
typedef float v8f __attribute__((ext_vector_type(8)));

// -------- wave32 sum via V_WMMA_F32_16X16X4_F32 (CDNA5 matrix pipe) --------
// A (16x4 f32, 2 VGPRs): lanes 0-15 hold K=0 in v0, lanes 16-31 hold K=2 in v0,
// v1 zeroed.  B = all-ones.  D[m][n] = partial_m + partial_{m+16} = S_m.
// D layout: VGPR j -> M=j (lanes 0-15) / M=8+j (lanes 16-31), so summing the 8
// components gives sum(S_0..7) on lanes<16 and sum(S_8..15) on lanes>=16;
// one xor-16 shuffle completes the full 32-lane sum (broadcast to all lanes).
// NOTE: call sites must be in uniform control flow (EXEC all ones).
__device__ __forceinline__ float wave_sum_wmma(float x) {
  v2f a; a[0] = x;   a[1] = 0.0f;
  v2f b; b[0] = 1.0f; b[1] = 1.0f;
  v8f c = {};
  v8f d = __builtin_amdgcn_wmma_f32_16x16x4_f32(
      /*neg_a=*/false, a, /*neg_b=*/false, b,
      /*c_mod=*/(short)0, c, /*reuse_a=*/false, /*reuse_b=*/false);
  float s = d[0] + d[1] + d[2] + d[3] + d[4] + d[5] + d[6] + d[7];
  s += __shfl_xor(s, 16, 32);
  return s;
}

// ---------------- scalar helpers ----------------
__device__ __forceinline__ float sigmoid_(float x) { return 1.0f / (1.0f + expf(-x)); }

__device__ __forceinline__ float bce_(float x, float t) {
  // max(x,0) - x*t + log1p(exp(-|x|))
  return fmaxf(x, 0.0f) - x * t + log1pf(expf(-fabsf(x)));
}

__device__ __forceinline__ float ciou_(float x1, float y1, float w1, float h1,
                                       float x2, float y2, float w2, float h2) {
  const float eps = 1e-7f;
  float b1x1 = x1 - 0.5f * w1, b1x2 = x1 + 0.5f * w1;
  float b1y1 = y1 - 0.5f * h1, b1y2 = y1 + 0.5f * h1;
  float b2x1 = x2 - 0.5f * w2, b2x2 = x2 + 0.5f * w2;
  float b2y1 = y2 - 0.5f * h2, b2y2 = y2 + 0.5f * h2;
  float iw = fmaxf(fminf(b1x2, b2x2) - fmaxf(b1x1, b2x1), 0.0f);
  float ih = fmaxf(fminf(b1y2, b2y2) - fmaxf(b1y1, b2y1), 0.0f);
  float inter = iw * ih;
  float uni = w1 * h1 + w2 * h2 - inter + eps;
  float iou = inter / uni;
  float cw = fmaxf(b1x2, b2x2) - fminf(b1x1, b2x1);
  float ch = fmaxf(b1y2, b2y2) - fminf(b1y1, b2y1);
  float c2 = cw * cw + ch * ch + eps;
  float dx = b2x1 + b2x2 - b1x1 - b1x2;
  float dy = b2y1 + b2y2 - b1y1 - b1y2;
  float rho2 = (dx * dx + dy * dy) * 0.25f;
  float da = atanf(w2 / (h2 + eps)) - atanf(w1 / (h1 + eps));
  const float k = 4.0f / (3.14159265358979323846f * 3.14159265358979323846f);
  float v = k * da * da;
  float alpha = v / (v - iou + (1.0f + eps));   // stop_gradient: forward identical
  return iou - (rho2 / c2 + v * alpha);
}

// ---------------- kernel 0: zero workspace ----------------
__global__ void zero_ws_kernel(float* __restrict__ ws, int n) {
  int stride = gridDim.x * blockDim.x;
  for (int i = blockIdx.x * blockDim.x + threadIdx.x; i < n; i += stride)
    ws[i] = 0.0f;
}

// ---------------- kernel 1: per-level target losses + tobj scatter ----------------
__global__ void targets_kernel(const float* __restrict__ pi,
                               const float* __restrict__ kpt,
                               float* __restrict__ tobj,
                               float* __restrict__ sums,   // 5 floats
                               int W,
                               float aw0, float ah0, float aw1, float ah1,
                               float aw2, float ah2) {
  int r = blockIdx.x * blockDim.x + threadIdx.x;
  float c_cnt = 0.0f, c_box = 0.0f, c_kptv = 0.0f, c_kpt = 0.0f, c_np = 0.0f;

  if (r < ROWS_) {
    int o   = r / (NA_ * NT_);
    int rem = r - o * (NA_ * NT_);
    int a   = rem / NT_;
    int n   = rem - a * NT_;

    const float* t = kpt + n * D0_;
    float Wf = (float)W;                       // H == W at every level
    float gx = t[2] * Wf, gy = t[3] * Wf;
    float gw = t[4] * Wf, gh = t[5] * Wf;

    float aw = (a == 0) ? aw0 : ((a == 1) ? aw1 : aw2);
    float ah = (a == 0) ? ah0 : ((a == 1) ? ah1 : ah2);

    float r1 = gw / aw, r2 = gh / ah;
    bool m_anchor =
        fmaxf(fmaxf(r1, 1.0f / r1), fmaxf(r2, 1.0f / r2)) < 4.0f;

    bool sel = false;
    float ox = 0.0f, oy = 0.0f;
    switch (o) {
      case 0: sel = true; break;
      case 1: sel = (fmodf(gx, 1.0f) < 0.5f) && (gx > 1.0f); ox =  0.5f; break;
      case 2: sel = (fmodf(gy, 1.0f) < 0.5f) && (gy > 1.0f); oy =  0.5f; break;
      case 3: { float gxi = Wf - gx;
                sel = (fmodf(gxi, 1.0f) < 0.5f) && (gxi > 1.0f); ox = -0.5f; } break;
      default:{ float gyi = Wf - gy;
                sel = (fmodf(gyi, 1.0f) < 0.5f) && (gyi > 1.0f); oy = -0.5f; } break;
    }

    if (sel && m_anchor) {
      int gi = (int)(gx - ox); gi = min(max(gi, 0), W - 1);   // trunc + clip
      int gj = (int)(gy - oy); gj = min(max(gj, 0), W - 1);
      float fgi = (float)gi, fgj = (float)gj;
      int b = (int)t[0];

      size_t cell = (size_t)(((b * NA_ + a) * W + gj) * W + gi);
      const float* ps = pi + cell * (size_t)NO_;

      // box loss
      float px = sigmoid_(ps[0]) * 2.0f - 0.5f;
      float py = sigmoid_(ps[1]) * 2.0f - 0.5f;
      float sw = sigmoid_(ps[2]) * 2.0f; float pw = sw * sw * aw;
      float sh = sigmoid_(ps[3]) * 2.0f; float ph = sh * sh * ah;
      float iou = ciou_(px, py, pw, ph, gx - fgi, gy - fgj, gw, gh);
      c_box = 1.0f - iou;
      c_cnt = 1.0f;

      // objectness target scatter: val >= 0, atomicMax on raw bits is an
      // order-independent (deterministic) resolution of duplicate indices.
      float val = fmaxf(iou, 0.0f);            // GR==1 -> 1-GR+GR*clip(iou,0)
      atomicMax((unsigned int*)(tobj + cell), __float_as_uint(val));

      // keypoints
      #pragma unroll
      for (int k = 0; k < NKPT_; ++k) {
        float pkx = ps[6 + 3 * k] * 2.0f - 0.5f;   // no sigmoid (reference)
        float pky = ps[7 + 3 * k] * 2.0f - 0.5f;
        float pks = ps[8 + 3 * k];
        float tx0 = t[6 + 2 * k] * Wf;
        float ty0 = t[7 + 2 * k] * Wf;
        float sx = (tx0 != 0.0f) ? tx0 - fgi : tx0;
        float sy = (ty0 != 0.0f) ? ty0 - fgj : ty0;
        bool km = (sx != 0.0f);
        c_kptv += bce_(pks, km ? 1.0f : 0.0f);
        if (km) {
          float dx = pkx - sx, dy = pky - sy;
          float d = dx * dx + dy * dy;
          c_kpt += logf(d + 1.0f + 1e-9f);
          c_np  += 1.0f;
        }
      }
    }
  }

  // uniform control flow from here: every thread (incl. r>=ROWS_) participates
  float s0 = wave_sum_wmma(c_cnt);
  float s1 = wave_sum_wmma(c_box);
  float s2 = wave_sum_wmma(c_kptv);
  float s3 = wave_sum_wmma(c_kpt);
  float s4 = wave_sum_wmma(c_np);
  if ((threadIdx.x & 31) == 0) {
    atomicAdd(sums + 0, s0);
    atomicAdd(sums + 1, s1);
    atomicAdd(sums + 2, s2);
    atomicAdd(sums + 3, s3);
    atomicAdd(sums + 4, s4);
  }
}

// ---------------- kernel 2: objectness BCE sum (HBM streaming) ----------------
__global__ void obj_kernel(const float* __restrict__ pi,
                           const float* __restrict__ tobj,
                           int n, float* __restrict__ sumPtr) {
  float acc = 0.0f;
  int stride = gridDim.x * blockDim.x;
  for (int i = blockIdx.x * blockDim.x + threadIdx.x; i < n; i += stride) {
    __builtin_prefetch(pi + (size_t)(i + stride) * NO_ + 4, 0, 0);  // global_prefetch_b8
    float x = pi[(size_t)i * NO_ + 4];
    float t = tobj[i];
    acc += bce_(x, t);
  }
  float s = wave_sum_wmma(acc);   // uniform: grid-stride loop exits uniformly per wave? no-
  if ((threadIdx.x & 31) == 0) atomicAdd(sumPtr, s);
}

// ---------------- kernel 3: finalize ----------------
__global__ void finalize_kernel(const float* __restrict__ ws, float* __restrict__ out) {
  const float balance[3] = {4.0f, 1.0f, 0.4f};
  const float cells[3]   = {(float)(BS_ * NA_ * 80 * 80),
                            (float)(BS_ * NA_ * 40 * 40),
                            (float)(BS_ * NA_ * 20 * 20)};
  float lbox = 0.0f, lobj = 0.0f, lkpt = 0.0f, lkptv = 0.0f;
  for (int i = 0; i < 3; ++i) {
    const float* s = ws + i * 5;
    float cnt  = fmaxf(s[0], 1.0f);
    float npos = fmaxf(s[4], 1.0f);
    lbox  += s[1] / cnt;
    lkptv += s[2] / (cnt * (float)NKPT_);
    lkpt  += s[3] / npos;                 // factor*sum/(cnt*NKPT) == sum/npos
    lobj  += (ws[15 + i] / cells[i]) * balance[i];
  }
  out[0] = 0.05f * lbox + 0.7f * lobj + 0.1f * lkpt + 0.6f * lkptv;
}

// ---------------- host launcher ----------------
extern "C" void kernel_launch(void* const* d_in, const int* in_sizes, int n_in,
                              void* d_out, int out_size, void* d_ws, size_t ws_size,
                              hipStream_t stream) {
  const float* p0  = (const float*)d_in[0];
  const float* p1  = (const float*)d_in[1];
  const float* p2  = (const float*)d_in[2];
  const float* kpt = (const float*)d_in[3];
  float* ws  = (float*)d_ws;
  float* out = (float*)d_out;

  const int c0 = BS_ * NA_ * 80 * 80;
  const int c1 = BS_ * NA_ * 40 * 40;
  const int c2 = BS_ * NA_ * 20 * 20;
  float* tobj0 = ws + WS_HDR_;
  float* tobj1 = tobj0 + c0;
  float* tobj2 = tobj1 + c1;

  // 0) zero header + tobj (~1.6 MB of workspace)
  int nz = WS_HDR_ + c0 + c1 + c2;
  zero_ws_kernel<<<(nz + 255) / 256, 256, 0, stream>>>(ws, nz);

  // 1) target losses + tobj scatter, per level (anchors pre-divided by stride)
  const int tb = (ROWS_ + 255) / 256;
  targets_kernel<<<tb, 256, 0, stream>>>(p0, kpt, tobj0, ws + 0, 80,
      2.375f, 3.375f, 5.5f, 5.0f, 4.75f, 11.75f);
  targets_kernel<<<tb, 256, 0, stream>>>(p1, kpt, tobj1, ws + 5, 40,
      6.0f, 4.25f, 5.375f, 9.5f, 11.25f, 8.5625f);
  targets_kernel<<<tb, 256, 0, stream>>>(p2, kpt, tobj2, ws + 10, 20,
      4.375f, 9.40625f, 9.46875f, 8.25f, 7.4375f, 16.9375f);

  // 2) objectness BCE sums
  auto blocks = [](int n) { int b = (n + 255) / 256; return b > 1024 ? 1024 : b; };
  obj_kernel<<<blocks(c0), 256, 0, stream>>>(p0, tobj0, c0, ws + 15);
  obj_kernel<<<blocks(c1), 256, 0, stream>>>(p1, tobj1, c1, ws + 16);
  obj_kernel<<<blocks(c2), 256, 0, stream>>>(p2, tobj2, c2, ws + 17);

  // 3) combine into scalar loss
  finalize_kernel<<<1, 1, 0, stream>>>(ws, out);
}